// EEGModel_18940805775671
// MI455X (gfx1250) — compile-verified
//
#include <hip/hip_runtime.h>
#include <math.h>

typedef __attribute__((ext_vector_type(2))) float v2f;
typedef __attribute__((ext_vector_type(8))) float v8f;

#define NB      70      // batch
#define TT      768     // time
#define CC      19      // channels
#define NCNN    70
#define FF      64
#define HH      128
#define G4      512     // 4*H
#define KPAD    60      // 57 padded to multiple of 4
#define NKCH    15      // k-chunks of 4 covering 57
#define EPSBN   1e-5f

static __device__ __forceinline__ v8f wmma4(v2f a, v2f b, v8f c) {
  // D = A(16x4,f32) * B(4x16,f32) + C(16x16,f32)   -> v_wmma_f32_16x16x4_f32
  return __builtin_amdgcn_wmma_f32_16x16x4_f32(false, a, false, b, (short)0, c, false, false);
}
static __device__ __forceinline__ v8f vzero8() {
  v8f z = {0.f,0.f,0.f,0.f,0.f,0.f,0.f,0.f};
  return z;
}
static __device__ __forceinline__ float sigmoidf_(float x) {
  return 1.0f / (1.0f + __expf(-x));
}

// ---------------------------------------------------------------------------
// Fused conv1d(K=3, pad=1) + bias + BN + ReLU + mean over T.
// One block (1 wave32) per (n, cnn). Both operands staged in LDS:
//   wlds: 60x64 zero-padded filter block (loaded once)
//   xlds: 18x19 x-tile (rows t0-1..t0+16, re-staged per t-tile)
// Per-lane A offsets/validity precomputed once -> hot loop is pure
// ds_load + v_cndmask + v_wmma, no divergence, no integer division.
// ---------------------------------------------------------------------------
__global__ __launch_bounds__(32) void conv_feats_kernel(
    const float* __restrict__ x,       // (70,768,19)
    const float* __restrict__ w,       // (70,64,19,3)
    const float* __restrict__ cb,      // (70,64)
    const float* __restrict__ gamma,   // (70,64)
    const float* __restrict__ beta,
    const float* __restrict__ mean,
    const float* __restrict__ var,
    float* __restrict__ feats)         // (70,70,64)
{
  const int n    = blockIdx.x;
  const int cnn  = blockIdx.y;
  const int lane = threadIdx.x;
  const int lm   = lane & 15;          // row/col within tile
  const int kh   = (lane >> 4) << 1;   // K sub-offset for this half-wave

  __shared__ float wlds[KPAD * FF];    // [k][f], 15 KB
  __shared__ float xlds[18 * CC];      // [r][c], r -> t0-1+r

  // stage weights (zero-padded K) into LDS
  for (int idx = lane; idx < KPAD * FF; idx += 32) {
    int k = idx / FF, f = idx % FF;
    float v = 0.f;
    if (k < 57) {
      int c = k / 3, j = k % 3;
      v = w[(((size_t)cnn * FF + f) * CC + c) * 3 + j];
    }
    wlds[idx] = v;
  }

  // per-lane affine constants for the 4 f-tiles (col = tile*16 + lm)
  float scl[4], sft[4];
#pragma unroll
  for (int t4 = 0; t4 < 4; ++t4) {
    int col = cnn * FF + t4 * 16 + lm;
    float s = gamma[col] * __frsqrt_rn(var[col] + EPSBN);
    scl[t4] = s;
    sft[t4] = (cb[col] - mean[col]) * s + beta[col];
  }

  // per-lane A-fragment LDS offsets (j*19+c) and validity mask, once
  int aoff[NKCH][2];
  unsigned amask = 0u;
#pragma unroll
  for (int i = 0; i < NKCH; ++i) {
#pragma unroll
    for (int v = 0; v < 2; ++v) {
      int k = i * 4 + kh + v;
      int o = 0;
      if (k < 57) {
        int c = k / 3, j = k - c * 3;
        o = j * CC + c;
        amask |= 1u << (i * 2 + v);
      }
      aoff[i][v] = o;
    }
  }
  const int lmrow = lm * CC;

  float colsum[4] = {0.f, 0.f, 0.f, 0.f};
  const float* xn = x + (size_t)n * TT * CC;
  __syncthreads();

  for (int t0 = 0; t0 < TT; t0 += 16) {
    // stage x tile: rows t0-1 .. t0+16 (zero at boundaries)
#pragma unroll
    for (int r = 0; r < 18; ++r) {
      if (lane < CC) {
        int tt = t0 - 1 + r;
        xlds[r * CC + lane] = (tt >= 0 && tt < TT) ? xn[tt * CC + lane] : 0.f;
      }
    }
    __syncthreads();

    v8f acc[4];
#pragma unroll
    for (int t4 = 0; t4 < 4; ++t4) acc[t4] = vzero8();

#pragma unroll
    for (int i = 0; i < NKCH; ++i) {
      v2f a;
      {
        float a0 = xlds[lmrow + aoff[i][0]];
        float a1 = xlds[lmrow + aoff[i][1]];
        a[0] = (amask & (1u << (i * 2))) ? a0 : 0.f;
        a[1] = (amask & (2u << (i * 2))) ? a1 : 0.f;
      }
      const int bbase = (i * 4 + kh) * FF + lm;
#pragma unroll
      for (int t4 = 0; t4 < 4; ++t4) {
        v2f b;
        b[0] = wlds[bbase + t4 * 16];
        b[1] = wlds[bbase + FF + t4 * 16];
        acc[t4] = wmma4(a, b, acc[t4]);
      }
    }

    // epilogue: affine + relu + row-sum into per-column accumulator
#pragma unroll
    for (int t4 = 0; t4 < 4; ++t4) {
      float s = 0.f;
#pragma unroll
      for (int r = 0; r < 8; ++r) {
        float y = acc[t4][r] * scl[t4] + sft[t4];
        s += fmaxf(y, 0.f);
      }
      colsum[t4] += s;
    }
    __syncthreads();
  }

  // combine the two half-wave row groups, write result
#pragma unroll
  for (int t4 = 0; t4 < 4; ++t4) {
    float tot = colsum[t4] + __shfl_xor(colsum[t4], 16, 32);
    if (lane < 16)
      feats[((size_t)n * NCNN + cnn) * FF + t4 * 16 + lane] = tot * (1.0f / TT);
  }
}

// ---------------------------------------------------------------------------
// Generic fp32 WMMA GEMM:  C(Mtot, N) = A @ B^T + bias,  B is (N, K) row-major.
// A rows use grouped addressing: row m -> (m/rowsPerGroup)*groupStride +
// (m%rowsPerGroup)*rowStride (handles strided feats slices). K % 4 == 0.
// Block = 128 threads (4 waves); wave handles a 16x64 strip. Fragment loads
// are branchless 8-byte vector loads (clamped address + cndmask on the edge).
// ---------------------------------------------------------------------------
__global__ __launch_bounds__(128) void gemm_bias_kernel(
    const float* __restrict__ A, const float* __restrict__ B,
    const float* __restrict__ bias, float* __restrict__ C,
    int Mtot, int N, int K, int rowsPerGroup, int rowStride, int groupStride,
    int ldc)
{
  const int wave = threadIdx.x >> 5;
  const int lane = threadIdx.x & 31;
  const int lm   = lane & 15;
  const int kh   = (lane >> 4) << 1;
  const int tileM   = blockIdx.x;
  const int colBase = blockIdx.y * 256 + wave * 64;
  if (colBase >= N) return;

  const int  mA  = tileM * 16 + lm;
  const bool aOK = (mA < Mtot);
  const int  mC  = aOK ? mA : 0;
  const size_t aRow =
      (size_t)(mC / rowsPerGroup) * groupStride + (size_t)(mC % rowsPerGroup) * rowStride;

  v8f acc[4];
#pragma unroll
  for (int t4 = 0; t4 < 4; ++t4) acc[t4] = vzero8();

  for (int k0 = 0; k0 < K; k0 += 4) {
    v2f a = *(const v2f*)(A + aRow + k0 + kh);   // 8-byte aligned (even offsets)
    a[0] = aOK ? a[0] : 0.f;
    a[1] = aOK ? a[1] : 0.f;
#pragma unroll
    for (int t4 = 0; t4 < 4; ++t4) {
      int nn = colBase + t4 * 16 + lm;
      v2f b = *(const v2f*)(B + (size_t)nn * K + k0 + kh);
      acc[t4] = wmma4(a, b, acc[t4]);
    }
  }
#pragma unroll
  for (int t4 = 0; t4 < 4; ++t4) {
    int col = colBase + t4 * 16 + lm;
    float bv = bias ? bias[col] : 0.f;
#pragma unroll
    for (int r = 0; r < 8; ++r) {
      int m = tileM * 16 + r + ((lane >> 4) << 3);
      if (m < Mtot) C[(size_t)m * ldc + col] = acc[t4][r] + bv;
    }
  }
}

// ---------------------------------------------------------------------------
// Persistent BiLSTM layer scan. 4 blocks: (lstm1,f),(lstm1,r),(lstm2,f),(lstm2,r).
// Per step: gates(70,512) = h @ Whh^T + xg[t]  via WMMA (160 tiles / 16 waves),
// then cell update. Work buffers (c,h,gates per sequence) in global scratch.
// ---------------------------------------------------------------------------
__global__ __launch_bounds__(512) void lstm_scan_kernel(
    const float* __restrict__ xgA, const float* __restrict__ xgB,   // (2,70*T,512)
    const float* __restrict__ whhA, const float* __restrict__ whhB, // (2,512,128)
    float* __restrict__ hsA, float* __restrict__ hsB,               // (70,T,256)
    float* __restrict__ work, int TA, int TB)
{
  const int seq  = blockIdx.x;
  const int lstm = seq >> 1;
  const int dir  = seq & 1;
  const int T    = lstm ? TB : TA;
  const float* xg  = (lstm ? xgB : xgA) + (size_t)dir * NB * T * G4;
  const float* whh = (lstm ? whhB : whhA) + (size_t)dir * G4 * HH;
  float* hseq = lstm ? hsB : hsA;
  const int colOff = dir * HH;

  const size_t wstride = (size_t)NB * HH * 2 + (size_t)NB * G4; // c + h + gates
  float* cbuf  = work + (size_t)seq * wstride;
  float* hbuf  = cbuf + NB * HH;
  float* gates = hbuf + NB * HH;

  const int tid  = threadIdx.x;
  const int wave = tid >> 5;
  const int lane = tid & 31;
  const int lm   = lane & 15;
  const int kh   = (lane >> 4) << 1;

  for (int e = tid; e < NB * HH; e += 512) { cbuf[e] = 0.f; hbuf[e] = 0.f; }
  __threadfence();
  __syncthreads();

  for (int step = 0; step < T; ++step) {
    const int t = dir ? (T - 1 - step) : step;

    // gates = h @ Whh^T + xg[:, t, :]
#pragma unroll 1
    for (int i = 0; i < 10; ++i) {
      const int tile = wave + i * 16;            // 160 tiles total
      const int tM = tile >> 5, tN = tile & 31;
      const int  mA  = tM * 16 + lm;
      const bool aOK = (mA < NB);
      const int  mR  = aOK ? mA : 0;
      const float* hrow = hbuf + mR * HH + kh;
      const float* brow = whh + (size_t)(tN * 16 + lm) * HH + kh;
      v8f acc = vzero8();
      for (int k0 = 0; k0 < HH; k0 += 4) {
        v2f a = *(const v2f*)(hrow + k0);
        a[0] = aOK ? a[0] : 0.f;
        a[1] = aOK ? a[1] : 0.f;
        v2f b = *(const v2f*)(brow + k0);
        acc = wmma4(a, b, acc);
      }
      const int col = tN * 16 + lm;
#pragma unroll
      for (int r = 0; r < 8; ++r) {
        int m = tM * 16 + r + ((lane >> 4) << 3);
        if (m < NB)
          gates[m * G4 + col] = acc[r] + xg[((size_t)m * T + t) * G4 + col];
      }
    }
    __threadfence();
    __syncthreads();

    // cell update
    for (int e = tid; e < NB * HH; e += 512) {
      int b_ = e >> 7, hh = e & (HH - 1);
      int base = b_ * G4 + hh;
      float ig = sigmoidf_(gates[base]);
      float fg = sigmoidf_(gates[base + HH]);
      float gg = tanhf(gates[base + 2 * HH]);
      float og = sigmoidf_(gates[base + 3 * HH]);
      float cn = fg * cbuf[e] + ig * gg;
      float hn = og * tanhf(cn);
      cbuf[e] = cn;
      hbuf[e] = hn;
      hseq[((size_t)b_ * T + t) * 256 + colOff + hh] = hn;
    }
    __threadfence();
    __syncthreads();
  }
}

// ---------------------------------------------------------------------------
// combined(70,512) = [ mean_t h1seq_l1 , mean_t h1seq_l2 ]
// ---------------------------------------------------------------------------
__global__ void mean_kernel(const float* __restrict__ h1,
                            const float* __restrict__ h2,
                            float* __restrict__ comb)
{
  int idx = blockIdx.x * blockDim.x + threadIdx.x;
  if (idx >= NB * 512) return;
  int n = idx >> 9, col = idx & 511;
  float s = 0.f;
  if (col < 256) {
    for (int t = 0; t < 30; ++t) s += h1[((size_t)n * 30 + t) * 256 + col];
    s *= (1.0f / 30.0f);
  } else {
    for (int t = 0; t < 40; ++t) s += h2[((size_t)n * 40 + t) * 256 + (col - 256)];
    s *= (1.0f / 40.0f);
  }
  comb[idx] = s;
}

// ---------------------------------------------------------------------------
// head: fc1(512->64)+relu, fc2(64->1)+sigmoid
// ---------------------------------------------------------------------------
__global__ __launch_bounds__(64) void head_kernel(
    const float* __restrict__ comb, const float* __restrict__ fc1w,
    const float* __restrict__ fc1b, const float* __restrict__ fc2w,
    const float* __restrict__ fc2b, float* __restrict__ out)
{
  __shared__ float h[64];
  int n = blockIdx.x, j = threadIdx.x;
  float s = fc1b[j];
  const float* c = comb + (size_t)n * 512;
  for (int k = 0; k < 512; ++k) s += c[k] * fc1w[(size_t)j * 512 + k];
  h[j] = fmaxf(s, 0.f);
  __syncthreads();
  if (j == 0) {
    float s2 = fc2b[0];
    for (int k = 0; k < 64; ++k) s2 += h[k] * fc2w[k];
    out[n] = sigmoidf_(s2);
  }
}

// ---------------------------------------------------------------------------
extern "C" void kernel_launch(void* const* d_in, const int* in_sizes, int n_in,
                              void* d_out, int out_size, void* d_ws, size_t ws_size,
                              hipStream_t stream) {
  const float* x       = (const float*)d_in[0];
  const float* conv_w  = (const float*)d_in[1];
  const float* conv_b  = (const float*)d_in[2];
  const float* bn_g    = (const float*)d_in[3];
  const float* bn_b    = (const float*)d_in[4];
  const float* bn_m    = (const float*)d_in[5];
  const float* bn_v    = (const float*)d_in[6];
  const float* l1_wih0 = (const float*)d_in[7];
  const float* l1_whh0 = (const float*)d_in[8];
  const float* l1_b0   = (const float*)d_in[9];
  const float* l1_wih1 = (const float*)d_in[10];
  const float* l1_whh1 = (const float*)d_in[11];
  const float* l1_b1   = (const float*)d_in[12];
  const float* l2_wih0 = (const float*)d_in[13];
  const float* l2_whh0 = (const float*)d_in[14];
  const float* l2_b0   = (const float*)d_in[15];
  const float* l2_wih1 = (const float*)d_in[16];
  const float* l2_whh1 = (const float*)d_in[17];
  const float* l2_b1   = (const float*)d_in[18];
  const float* fc1_w   = (const float*)d_in[19];
  const float* fc1_b   = (const float*)d_in[20];
  const float* fc2_w   = (const float*)d_in[21];
  const float* fc2_b   = (const float*)d_in[22];
  float* out = (float*)d_out;
  float* ws  = (float*)d_ws;

  // workspace layout (floats)
  float* feats   = ws;                                 // 70*70*64      = 313600
  float* xg0_1   = feats   + 313600;                   // 2*2100*512    = 2150400
  float* h0s_1   = xg0_1   + 2150400;                  // 2100*256      = 537600
  float* xg1_1   = h0s_1   + 537600;                   // 2150400
  float* h1s_1   = xg1_1   + 2150400;                  // 537600
  float* xg0_2   = h1s_1   + 537600;                   // 2*2800*512    = 2867200
  float* h0s_2   = xg0_2   + 2867200;                  // 2800*256      = 716800
  float* xg1_2   = h0s_2   + 716800;                   // 2867200
  float* h1s_2   = xg1_2   + 2867200;                  // 716800
  float* scanwk  = h1s_2   + 716800;                   // 4*(8960+8960+35840)
  float* comb    = scanwk  + 4 * (NB*HH*2 + NB*G4);    // 70*512

  // 1) fused conv + BN + ReLU + time-mean
  conv_feats_kernel<<<dim3(NB, NCNN), 32, 0, stream>>>(
      x, conv_w, conv_b, bn_g, bn_b, bn_m, bn_v, feats);

  // 2) layer-0 input gates: xg = feats_slice @ Wih0^T + b0  (per lstm, per dir)
  for (int d = 0; d < 2; ++d) {
    gemm_bias_kernel<<<dim3(132, 2), 128, 0, stream>>>(
        feats, l1_wih0 + (size_t)d * G4 * FF, l1_b0 + d * G4,
        xg0_1 + (size_t)d * 2100 * G4, 2100, G4, FF, 30, FF, NCNN * FF, G4);
    gemm_bias_kernel<<<dim3(175, 2), 128, 0, stream>>>(
        feats + 30 * FF, l2_wih0 + (size_t)d * G4 * FF, l2_b0 + d * G4,
        xg0_2 + (size_t)d * 2800 * G4, 2800, G4, FF, 40, FF, NCNN * FF, G4);
  }

  // 3) layer-0 scans (4 independent sequences, persistent blocks)
  lstm_scan_kernel<<<4, 512, 0, stream>>>(
      xg0_1, xg0_2, l1_whh0, l2_whh0, h0s_1, h0s_2, scanwk, 30, 40);

  // 4) layer-1 input gates from concatenated layer-0 outputs (K=256)
  for (int d = 0; d < 2; ++d) {
    gemm_bias_kernel<<<dim3(132, 2), 128, 0, stream>>>(
        h0s_1, l1_wih1 + (size_t)d * G4 * 256, l1_b1 + d * G4,
        xg1_1 + (size_t)d * 2100 * G4, 2100, G4, 256, 30, 256, 30 * 256, G4);
    gemm_bias_kernel<<<dim3(175, 2), 128, 0, stream>>>(
        h0s_2, l2_wih1 + (size_t)d * G4 * 256, l2_b1 + d * G4,
        xg1_2 + (size_t)d * 2800 * G4, 2800, G4, 256, 40, 256, 40 * 256, G4);
  }

  // 5) layer-1 scans
  lstm_scan_kernel<<<4, 512, 0, stream>>>(
      xg1_1, xg1_2, l1_whh1, l2_whh1, h1s_1, h1s_2, scanwk, 30, 40);

  // 6) time-mean + concat
  mean_kernel<<<(NB * 512 + 255) / 256, 256, 0, stream>>>(h1s_1, h1s_2, comb);

  // 7) FC head
  head_kernel<<<NB, 64, 0, stream>>>(comb, fc1_w, fc1_b, fc2_w, fc2_b, out);
}